// MLP_85598698209927
// MI455X (gfx1250) — compile-verified
//
#include <hip/hip_runtime.h>
#include <hip/hip_bf16.h>

// Problem sizes (fixed by the reference).
#define N_   1024
#define DIN  1024
#define DH   512

typedef __attribute__((ext_vector_type(2))) float v2f;
typedef __attribute__((ext_vector_type(8))) float v8f;

__device__ __forceinline__ float fast_tanh(float x) {
#if __has_builtin(__builtin_amdgcn_tanhf)
    return __builtin_amdgcn_tanhf(x);   // v_tanh_f32 on gfx1250
#else
    return tanhf(x);
#endif
}

// ---------------------------------------------------------------------------
// Phase 1: H = vectors @ W_h^T + b_h ; M = vectors @ W_m^T + b_m
// fp32 WMMA 16x16x4. One wave -> one 16x16 tile of BOTH H and M (shared A frag).
// Block = 4 waves, each wave a different 16-row slab. Grid = (1024/64, 512/16).
//
// A-frag (16x4 f32, ISA 7.12.2): lane<16 holds M=lane, K={k0,k0+1};
//                                lane>=16 holds M=lane-16, K={k0+2,k0+3}.
// B-frag (4x16): lane<16 holds N=lane, K={k0,k0+1}; lane>=16 N=lane-16, K={k0+2,k0+3}.
// B[k][n] = W[n][k] (both K-contiguous -> float2 loads).
// C/D (16x16 f32): VGPR v, lane l: row = v + 8*(l>=16), col = l%16.
// ---------------------------------------------------------------------------
__global__ __launch_bounds__(128) void proj_kernel(
    const float* __restrict__ vectors,
    const float* __restrict__ W_h, const float* __restrict__ b_h,
    const float* __restrict__ W_m, const float* __restrict__ b_m,
    float* __restrict__ H, float* __restrict__ M)
{
    const int lane = threadIdx.x & 31;
    const int wave = threadIdx.x >> 5;
    const int row_base = blockIdx.x * 64 + wave * 16;
    const int col_base = blockIdx.y * 16;
    const int half = lane >> 4;      // 0 or 1
    const int lid  = lane & 15;
    const int ar   = row_base + lid; // A row this lane feeds
    const int bc   = col_base + lid; // output column / W row this lane feeds
    const int koff = half * 2;

    const float* pa = vectors + (size_t)ar * DIN + koff;
    const float* ph = W_h     + (size_t)bc * DIN + koff;
    const float* pm = W_m     + (size_t)bc * DIN + koff;

    v8f acc_h = {};
    v8f acc_m = {};

    #pragma unroll 4
    for (int k = 0; k < DIN; k += 4) {
        v2f a  = *(const v2f*)pa;
        v2f bh = *(const v2f*)ph;
        v2f bm = *(const v2f*)pm;
        acc_h = __builtin_amdgcn_wmma_f32_16x16x4_f32(
            false, a, false, bh, (short)0, acc_h, false, false);
        acc_m = __builtin_amdgcn_wmma_f32_16x16x4_f32(
            false, a, false, bm, (short)0, acc_m, false, false);
        pa += 4; ph += 4; pm += 4;
    }

    const float bhv = b_h[bc];
    const float bmv = b_m[bc];
    #pragma unroll
    for (int v = 0; v < 8; ++v) {
        const int row = row_base + v + half * 8;
        H[(size_t)row * DH + bc] = acc_h[v] + bhv;
        M[(size_t)row * DH + bc] = acc_m[v] + bmv;
    }
}

// ---------------------------------------------------------------------------
// Phase 2: out[i][j] = b2 + sum_k w2[k] * tanh(H[i][k] + M[j][k])
// 32x32 output tile / 256-thread block; K chunked by 128 through LDS.
// Wave mapping: lane -> j (Ms stride-129 reads are bank-conflict-free),
// each wave owns 4 i rows (Hs/w2 reads broadcast), stores coalesced in j.
// ---------------------------------------------------------------------------
#define TI   32
#define TJ   32
#define KC   128
#define LSTR 129   // padded LDS row stride (conflict-free lane-strided reads)

__global__ __launch_bounds__(256) void pair_kernel(
    const float* __restrict__ H, const float* __restrict__ M,
    const float* __restrict__ w2, const float* __restrict__ b2,
    float* __restrict__ out)
{
    __shared__ float Hs[TI * LSTR];
    __shared__ float Ms[TJ * LSTR];
    __shared__ float w2s[DH];

    const int t      = threadIdx.x;
    const int i_base = blockIdx.y * TI;
    const int j_base = blockIdx.x * TJ;
    const int jj     = t & 31;          // lane -> j within tile
    const int ii0    = (t >> 5) * 4;    // wave -> 4 consecutive i's

    for (int k = t; k < DH; k += 256) w2s[k] = w2[k];

    float acc0 = 0.f, acc1 = 0.f, acc2 = 0.f, acc3 = 0.f;

    for (int kc = 0; kc < DH; kc += KC) {
        __syncthreads();
        // Stage chunk: 32 rows x 128 floats for H and M (float4 global loads).
        #pragma unroll
        for (int p = 0; p < 4; ++p) {
            const int lin = t + p * 256;        // 0..1023
            const int row = lin >> 5;           // 0..31
            const int k4  = (lin & 31) * 4;     // 0..124
            const float4 hv = *(const float4*)(H + (size_t)(i_base + row) * DH + kc + k4);
            const float4 mv = *(const float4*)(M + (size_t)(j_base + row) * DH + kc + k4);
            float* hd = &Hs[row * LSTR + k4];
            float* md = &Ms[row * LSTR + k4];
            hd[0] = hv.x; hd[1] = hv.y; hd[2] = hv.z; hd[3] = hv.w;
            md[0] = mv.x; md[1] = mv.y; md[2] = mv.z; md[3] = mv.w;
        }
        __syncthreads();

        // Hint the next chunk toward the caches while we crunch this one.
        if (kc + KC < DH) {
            __builtin_prefetch(H + (size_t)(i_base + (t & 31)) * DH + kc + KC, 0, 1);
            __builtin_prefetch(M + (size_t)(j_base + (t & 31)) * DH + kc + KC, 0, 1);
        }

        const float* hr0 = &Hs[(ii0 + 0) * LSTR];
        const float* hr1 = &Hs[(ii0 + 1) * LSTR];
        const float* hr2 = &Hs[(ii0 + 2) * LSTR];
        const float* hr3 = &Hs[(ii0 + 3) * LSTR];
        const float* mr  = &Ms[jj * LSTR];
        const float* wv  = &w2s[kc];

        #pragma unroll 4
        for (int k = 0; k < KC; ++k) {
            const float m = mr[k];
            const float w = wv[k];
            acc0 += w * fast_tanh(hr0[k] + m);
            acc1 += w * fast_tanh(hr1[k] + m);
            acc2 += w * fast_tanh(hr2[k] + m);
            acc3 += w * fast_tanh(hr3[k] + m);
        }
    }

    const float bias = b2[0];
    float* o = out + (size_t)(i_base + ii0) * N_ + j_base + jj;
    o[0 * N_] = acc0 + bias;
    o[1 * N_] = acc1 + bias;
    o[2 * N_] = acc2 + bias;
    o[3 * N_] = acc3 + bias;
}

// ---------------------------------------------------------------------------
extern "C" void kernel_launch(void* const* d_in, const int* in_sizes, int n_in,
                              void* d_out, int out_size, void* d_ws, size_t ws_size,
                              hipStream_t stream) {
    const float* vectors = (const float*)d_in[0];
    const float* W_h     = (const float*)d_in[1];
    const float* b_h     = (const float*)d_in[2];
    const float* W_m     = (const float*)d_in[3];
    const float* b_m     = (const float*)d_in[4];
    const float* w2      = (const float*)d_in[5];
    const float* b2      = (const float*)d_in[6];
    float* out = (float*)d_out;

    // Workspace: H (1024x512) followed by M (1024x512) = 4 MB fp32.
    float* Hbuf = (float*)d_ws;
    float* Mbuf = Hbuf + (size_t)N_ * DH;

    dim3 gp(N_ / 64, DH / 16);   // 16 x 32 blocks, 128 threads (4 waves)
    proj_kernel<<<gp, 128, 0, stream>>>(vectors, W_h, b_h, W_m, b_m, Hbuf, Mbuf);

    dim3 gq(N_ / TJ, N_ / TI);   // 32 x 32 blocks, 256 threads
    pair_kernel<<<gq, 256, 0, stream>>>(Hbuf, Mbuf, w2, b2, out);
}